// SimpleHAN_29575144800806
// MI455X (gfx1250) — compile-verified
//
#include <hip/hip_runtime.h>
#include <math.h>

// ---------------- problem constants (match reference) ----------------
#define NTX   100000
#define NW    50000
#define FTX   166
#define FW    64
#define HID_  64
#define HEADS_ 4
#define DH    16
#define ETT   1600000
#define ETW   800000
#define EWT   800000

typedef __attribute__((ext_vector_type(2))) float v2f;
typedef __attribute__((ext_vector_type(8))) float v8f;

// ---------------- WMMA f32 16x16x4 wrapper ----------------
// D = A(16x4) * B(4x16) + C(16x16), fp32 throughout (reference is fp32).
__device__ __forceinline__ v8f wmma4(v2f a, v2f b, v8f c) {
  return __builtin_amdgcn_wmma_f32_16x16x4_f32(
      /*neg_a=*/false, a, /*neg_b=*/false, b,
      /*c_mod=*/(short)0, c, /*reuse_a=*/false, /*reuse_b=*/false);
}

__device__ __forceinline__ float nan2num(float x) {
  if (isnan(x)) return 0.0f;
  if (isinf(x)) return x > 0.0f ? 3.4028234663852886e38f : -3.4028234663852886e38f;
  return x;
}

// Y[N,64] = (clean? nan_to_num(X) : X)[N,K] @ W[K,64] + bias[64]
// one wave per 16x64 tile; 4 waves (128 thr) per block -> 64 rows/block.
// A rows are clamped to N-1 (garbage only feeds C rows >= N, never stored),
// so the K-body runs with zero predication: one global_load_b64 for A per step.
template <bool CLEAN>
__global__ __launch_bounds__(128) void han_gemm_bias(
    const float* __restrict__ X, const float* __restrict__ W,
    const float* __restrict__ bias, float* __restrict__ Y, int N, int K) {
  const int lane = threadIdx.x & 31;
  const int wave = threadIdx.x >> 5;
  const int row0 = (blockIdx.x * 4 + wave) * 16;
  if (row0 >= N) return;
  const int m  = lane & 15;   // A-row / B,C-col within tile
  const int hi = lane >> 4;   // K-half for A/B, M-half for C
  const int arow = min(row0 + m, N - 1);
  const float* __restrict__ Xrow = X + (size_t)arow * K;

  v8f zero = {0.f, 0.f, 0.f, 0.f, 0.f, 0.f, 0.f, 0.f};
  v8f acc[4] = {zero, zero, zero, zero};

  int k0 = 0;
  for (; k0 + 4 <= K; k0 += 4) {
    const int ka = k0 + 2 * hi;
    v2f a = *(const v2f*)(Xrow + ka);  // K even, ka even -> 8B aligned
    if (CLEAN) { a.x = nan2num(a.x); a.y = nan2num(a.y); }
#pragma unroll
    for (int nt = 0; nt < 4; ++nt) {
      const int col = nt * 16 + m;
      v2f b;
      b.x = W[(size_t)ka * HID_ + col];
      b.y = W[(size_t)(ka + 1) * HID_ + col];
      acc[nt] = wmma4(a, b, acc[nt]);
    }
  }
  if (k0 < K) {  // K % 4 != 0 tail (only F_TX=166 path), predicated
    const int ka = k0 + 2 * hi;
    v2f a;
    a.x = (ka < K) ? Xrow[ka] : 0.f;
    a.y = (ka + 1 < K) ? Xrow[ka + 1] : 0.f;
    if (CLEAN) { a.x = nan2num(a.x); a.y = nan2num(a.y); }
#pragma unroll
    for (int nt = 0; nt < 4; ++nt) {
      const int col = nt * 16 + m;
      v2f b;
      b.x = (ka < K)     ? W[(size_t)ka * HID_ + col]       : 0.f;
      b.y = (ka + 1 < K) ? W[(size_t)(ka + 1) * HID_ + col] : 0.f;
      acc[nt] = wmma4(a, b, acc[nt]);
    }
  }
#pragma unroll
  for (int nt = 0; nt < 4; ++nt) {
    const int col = nt * 16 + m;
    const float bv = bias[col];
#pragma unroll
    for (int i = 0; i < 8; ++i) {
      const int r = row0 + hi * 8 + i;
      if (r < N) Y[(size_t)r * HID_ + col] = acc[nt][i] + bv;
    }
  }
}

// colsum[c] += sum_n tanh((X @ Wk + bk)[n,c])  (semantic attention mean-reduce)
__global__ __launch_bounds__(128) void han_tanh_colsum(
    const float* __restrict__ X, const float* __restrict__ W,
    const float* __restrict__ bias, float* __restrict__ colsum, int N) {
  const int lane = threadIdx.x & 31;
  const int wave = threadIdx.x >> 5;
  const int row0 = (blockIdx.x * 4 + wave) * 16;
  if (row0 >= N) return;
  const int m  = lane & 15;
  const int hi = lane >> 4;
  const int arow = min(row0 + m, N - 1);  // clamp; r<N guard below excludes garbage
  const float* __restrict__ Xrow = X + (size_t)arow * HID_;

  v8f zero = {0.f, 0.f, 0.f, 0.f, 0.f, 0.f, 0.f, 0.f};
  v8f acc[4] = {zero, zero, zero, zero};

  for (int k0 = 0; k0 < HID_; k0 += 4) {
    const int ka = k0 + 2 * hi;
    v2f a = *(const v2f*)(Xrow + ka);
#pragma unroll
    for (int nt = 0; nt < 4; ++nt) {
      const int col = nt * 16 + m;
      v2f b;
      b.x = W[(size_t)ka * HID_ + col];
      b.y = W[(size_t)(ka + 1) * HID_ + col];
      acc[nt] = wmma4(a, b, acc[nt]);
    }
  }
#pragma unroll
  for (int nt = 0; nt < 4; ++nt) {
    const int col = nt * 16 + m;
    const float bv = bias[col];
    float part = 0.f;
#pragma unroll
    for (int i = 0; i < 8; ++i) {
      const int r = row0 + hi * 8 + i;
      if (r < N) part += tanhf(acc[nt][i] + bv);
    }
    atomicAdd(&colsum[col], part);
  }
}

// ---------------- elementwise / edge kernels ----------------
__global__ void han_fill(float* __restrict__ p, float v, int n) {
  int i = blockIdx.x * blockDim.x + threadIdx.x;
  if (i < n) p[i] = v;
}

// as[n,h] = sum_d z[n, h*16+d] * att[h,d]
__global__ void han_alpha(const float* __restrict__ z, const float* __restrict__ att,
                          float* __restrict__ out, int N) {
  int idx = blockIdx.x * blockDim.x + threadIdx.x;
  if (idx >= N * HEADS_) return;
  int n = idx >> 2, h = idx & 3;
  float s = 0.f;
#pragma unroll
  for (int d = 0; d < DH; ++d)
    s += z[(size_t)n * HID_ + h * DH + d] * att[h * DH + d];
  out[idx] = s;
}

__device__ __forceinline__ void atomicMaxF(float* addr, float v) {
  if (v >= 0.f) atomicMax((int*)addr, __float_as_int(v));
  else          atomicMin((unsigned int*)addr, __float_as_uint(v));
}

__device__ __forceinline__ float leaky02(float a) { return a > 0.f ? a : 0.2f * a; }

__global__ void han_edge_max(const int* __restrict__ src, const int* __restrict__ dst,
                             const float* __restrict__ as_, const float* __restrict__ ad_,
                             float* __restrict__ amax, int E) {
  int idx = blockIdx.x * blockDim.x + threadIdx.x;
  if (idx >= E * HEADS_) return;
  int e = idx >> 2, h = idx & 3;
  int s = src[e], d = dst[e];
  float a = leaky02(as_[s * HEADS_ + h] + ad_[d * HEADS_ + h]);
  atomicMaxF(&amax[d * HEADS_ + h], a);
}

__global__ void han_fix_amax(float* __restrict__ amax, int n) {
  int i = blockIdx.x * blockDim.x + threadIdx.x;
  if (i >= n) return;
  float v = amax[i];
  amax[i] = isfinite(v) ? v : 0.f;   // empty segments -> 0 (matches reference)
}

__global__ void han_edge_den(const int* __restrict__ src, const int* __restrict__ dst,
                             const float* __restrict__ as_, const float* __restrict__ ad_,
                             const float* __restrict__ amax, float* __restrict__ den, int E) {
  int idx = blockIdx.x * blockDim.x + threadIdx.x;
  if (idx >= E * HEADS_) return;
  int e = idx >> 2, h = idx & 3;
  int s = src[e], d = dst[e];
  float a = leaky02(as_[s * HEADS_ + h] + ad_[d * HEADS_ + h]);
  atomicAdd(&den[d * HEADS_ + h], __expf(a - amax[d * HEADS_ + h]));
}

// out[dst, c] += z[src, c] * softmax_weight   (one thread per (edge, channel))
__global__ void han_edge_agg(const int* __restrict__ src, const int* __restrict__ dst,
                             const float* __restrict__ as_, const float* __restrict__ ad_,
                             const float* __restrict__ amax, const float* __restrict__ den,
                             const float* __restrict__ z, float* __restrict__ out, int E) {
  long long idx = (long long)blockIdx.x * blockDim.x + threadIdx.x;
  if (idx >= (long long)E * HID_) return;
  int e = (int)(idx >> 6);
  int c = (int)(idx & 63);
  int h = c >> 4;
  int s = src[e], d = dst[e];
  float a = leaky02(as_[s * HEADS_ + h] + ad_[d * HEADS_ + h]);
  float ex = __expf(a - amax[d * HEADS_ + h]);
  float w = ex / (den[d * HEADS_ + h] + 1e-16f);
  atomicAdd(&out[(size_t)d * HID_ + c], z[(size_t)s * HID_ + c] * w);
}

__global__ void han_relu_ip(float* __restrict__ p, int n) {
  int i = blockIdx.x * blockDim.x + threadIdx.x;
  if (i < n) p[i] = fmaxf(p[i], 0.f);
}

// scores over 2 metapaths -> softmax attn (single block of 64 threads)
__global__ void han_sem_attn(const float* __restrict__ colsum, const float* __restrict__ q,
                             float* __restrict__ attn, float invN) {
  __shared__ float sh0[64], sh1[64];
  int c = threadIdx.x;
  sh0[c] = q[c] * colsum[c] * invN;
  sh1[c] = q[c] * colsum[64 + c] * invN;
  __syncthreads();
  if (c == 0) {
    float s0 = 0.f, s1 = 0.f;
    for (int i = 0; i < 64; ++i) { s0 += sh0[i]; s1 += sh1[i]; }
    float mx = fmaxf(s0, s1);
    float e0 = __expf(s0 - mx), e1 = __expf(s1 - mx);
    float inv = 1.f / (e0 + e1);
    attn[0] = e0 * inv;
    attn[1] = e1 * inv;
  }
}

__global__ void han_combine2(const float* __restrict__ o0, const float* __restrict__ o1,
                             const float* __restrict__ attn, float* __restrict__ h, int n) {
  int i = blockIdx.x * blockDim.x + threadIdx.x;
  if (i >= n) return;
  float v = attn[0] * o0[i] + attn[1] * o1[i];
  h[i] = fmaxf(v, 0.f);
}

__global__ void han_copy_relu(const float* __restrict__ src, float* __restrict__ dstp, int n) {
  int i = blockIdx.x * blockDim.x + threadIdx.x;
  if (i < n) dstp[i] = fmaxf(src[i], 0.f);
}

__global__ void han_cls(const float* __restrict__ h, const float* __restrict__ w,
                        const float* __restrict__ b, float* __restrict__ out, int N) {
  int n = blockIdx.x * blockDim.x + threadIdx.x;
  if (n >= N) return;
  float s = b[0];
#pragma unroll
  for (int c = 0; c < HID_; ++c) s += h[(size_t)n * HID_ + c] * w[c];
  out[n] = s;
}

// ---------------- host orchestration ----------------
static inline int cdiv(long long a, long long b) { return (int)((a + b - 1) / b); }

extern "C" void kernel_launch(void* const* d_in, const int* in_sizes, int n_in,
                              void* d_out, int out_size, void* d_ws, size_t ws_size,
                              hipStream_t stream) {
  // Inputs flattened in setup_inputs() dict insertion order (tuples W,b in order):
  // 0 x_tx, 1 x_w, 2 edge_tt, 3 edge_tw, 4 edge_wt,
  // 5 lin_tx.W, 6 lin_tx.b, 7 lin_w.W, 8 lin_w.b,
  // per layer (13 arrays, base 9 and 22):
  //   +0 proj_tx.W +1 proj_tx.b +2 proj_w.W +3 proj_w.b
  //   +4 att_src_tt +5 att_dst_tt +6 att_src_tw +7 att_dst_tw +8 att_src_wt +9 att_dst_wt
  //   +10 k_lin.W +11 k_lin.b +12 q
  // 35 cls.W, 36 cls.b
  const float* x_tx = (const float*)d_in[0];
  const float* x_w  = (const float*)d_in[1];
  const int*   e_tt = (const int*)d_in[2];
  const int*   e_tw = (const int*)d_in[3];
  const int*   e_wt = (const int*)d_in[4];
  const float* lin_tx_W = (const float*)d_in[5];
  const float* lin_tx_b = (const float*)d_in[6];
  const float* lin_w_W  = (const float*)d_in[7];
  const float* lin_w_b  = (const float*)d_in[8];
  const float* cls_W = (const float*)d_in[35];
  const float* cls_b = (const float*)d_in[36];

  // workspace carve (fp32), ~154 MB total — fits inside MI455X's 192 MB L2
  float* p = (float*)d_ws;
  float* h_tx = p; p += (size_t)NTX * HID_;
  float* h_w  = p; p += (size_t)NW * HID_;
  float* z_tx = p; p += (size_t)NTX * HID_;
  float* z_w  = p; p += (size_t)NW * HID_;
  float* o_tt = p; p += (size_t)NTX * HID_;
  float* o_wt = p; p += (size_t)NTX * HID_;
  float* o_tw = p; p += (size_t)NW * HID_;
  float* as_tt = p; p += (size_t)NTX * HEADS_;
  float* ad_tt = p; p += (size_t)NTX * HEADS_;
  float* as_tw = p; p += (size_t)NTX * HEADS_;
  float* ad_tw = p; p += (size_t)NW * HEADS_;
  float* as_wt = p; p += (size_t)NW * HEADS_;
  float* ad_wt = p; p += (size_t)NTX * HEADS_;
  float* amax_tt = p; p += (size_t)NTX * HEADS_;
  float* den_tt  = p; p += (size_t)NTX * HEADS_;
  float* amax_tw = p; p += (size_t)NW * HEADS_;
  float* den_tw  = p; p += (size_t)NW * HEADS_;
  float* amax_wt = p; p += (size_t)NTX * HEADS_;
  float* den_wt  = p; p += (size_t)NTX * HEADS_;
  float* colsum = p; p += 128;
  float* attn   = p; p += 2;

  const int T = 256;
  const float NEG_INF = -__builtin_huge_valf();

  // input projections (with nan_to_num), via f32 WMMA
  han_gemm_bias<true><<<cdiv(NTX, 64), 128, 0, stream>>>(x_tx, lin_tx_W, lin_tx_b, h_tx, NTX, FTX);
  han_gemm_bias<true><<<cdiv(NW, 64), 128, 0, stream>>>(x_w, lin_w_W, lin_w_b, h_w, NW, FW);

  auto run_edges = [&](const int* edges, int E, const float* as_, const float* ad_,
                       const float* zsrc, float* amax, float* den, float* out, int n_dst) {
    const int* src = edges;
    const int* dst = edges + E;
    han_fill<<<cdiv((long long)n_dst * HEADS_, T), T, 0, stream>>>(amax, NEG_INF, n_dst * HEADS_);
    han_fill<<<cdiv((long long)n_dst * HEADS_, T), T, 0, stream>>>(den, 0.f, n_dst * HEADS_);
    han_fill<<<cdiv((long long)n_dst * HID_, T), T, 0, stream>>>(out, 0.f, n_dst * HID_);
    han_edge_max<<<cdiv((long long)E * HEADS_, T), T, 0, stream>>>(src, dst, as_, ad_, amax, E);
    han_fix_amax<<<cdiv((long long)n_dst * HEADS_, T), T, 0, stream>>>(amax, n_dst * HEADS_);
    han_edge_den<<<cdiv((long long)E * HEADS_, T), T, 0, stream>>>(src, dst, as_, ad_, amax, den, E);
    han_edge_agg<<<cdiv((long long)E * HID_, T), T, 0, stream>>>(src, dst, as_, ad_, amax, den, zsrc, out, E);
    han_relu_ip<<<cdiv((long long)n_dst * HID_, T), T, 0, stream>>>(out, n_dst * HID_);
  };

  for (int layer = 0; layer < 2; ++layer) {
    const int base = 9 + layer * 13;
    const float* ptxW = (const float*)d_in[base + 0];
    const float* ptxB = (const float*)d_in[base + 1];
    const float* pwW  = (const float*)d_in[base + 2];
    const float* pwB  = (const float*)d_in[base + 3];
    const float* a_s_tt = (const float*)d_in[base + 4];
    const float* a_d_tt = (const float*)d_in[base + 5];
    const float* a_s_tw = (const float*)d_in[base + 6];
    const float* a_d_tw = (const float*)d_in[base + 7];
    const float* a_s_wt = (const float*)d_in[base + 8];
    const float* a_d_wt = (const float*)d_in[base + 9];
    const float* kW = (const float*)d_in[base + 10];
    const float* kB = (const float*)d_in[base + 11];
    const float* qv = (const float*)d_in[base + 12];

    // per-type projections (WMMA)
    han_gemm_bias<false><<<cdiv(NTX, 64), 128, 0, stream>>>(h_tx, ptxW, ptxB, z_tx, NTX, HID_);
    han_gemm_bias<false><<<cdiv(NW, 64), 128, 0, stream>>>(h_w, pwW, pwB, z_w, NW, HID_);

    // node-level attention coefficients
    han_alpha<<<cdiv((long long)NTX * HEADS_, T), T, 0, stream>>>(z_tx, a_s_tt, as_tt, NTX);
    han_alpha<<<cdiv((long long)NTX * HEADS_, T), T, 0, stream>>>(z_tx, a_d_tt, ad_tt, NTX);
    han_alpha<<<cdiv((long long)NTX * HEADS_, T), T, 0, stream>>>(z_tx, a_s_tw, as_tw, NTX);
    han_alpha<<<cdiv((long long)NW * HEADS_, T), T, 0, stream>>>(z_w, a_d_tw, ad_tw, NW);
    han_alpha<<<cdiv((long long)NW * HEADS_, T), T, 0, stream>>>(z_w, a_s_wt, as_wt, NW);
    han_alpha<<<cdiv((long long)NTX * HEADS_, T), T, 0, stream>>>(z_tx, a_d_wt, ad_wt, NTX);

    // GAT softmax-aggregate per edge type
    run_edges(e_tt, ETT, as_tt, ad_tt, z_tx, amax_tt, den_tt, o_tt, NTX);
    run_edges(e_tw, ETW, as_tw, ad_tw, z_tx, amax_tw, den_tw, o_tw, NW);
    run_edges(e_wt, EWT, as_wt, ad_wt, z_w, amax_wt, den_wt, o_wt, NTX);

    // semantic attention for tx (2 metapaths); for w: softmax of 1 == identity
    han_fill<<<1, 128, 0, stream>>>(colsum, 0.f, 128);
    han_tanh_colsum<<<cdiv(NTX, 64), 128, 0, stream>>>(o_tt, kW, kB, colsum, NTX);
    han_tanh_colsum<<<cdiv(NTX, 64), 128, 0, stream>>>(o_wt, kW, kB, colsum + 64, NTX);
    han_sem_attn<<<1, 64, 0, stream>>>(colsum, qv, attn, 1.0f / (float)NTX);
    han_combine2<<<cdiv((long long)NTX * HID_, T), T, 0, stream>>>(o_tt, o_wt, attn, h_tx, NTX * HID_);
    han_copy_relu<<<cdiv((long long)NW * HID_, T), T, 0, stream>>>(o_tw, h_w, NW * HID_);
  }

  // classifier
  han_cls<<<cdiv(NTX, T), T, 0, stream>>>(h_tx, cls_W, cls_b, (float*)d_out, NTX);
}